// DualHypergraphAligner_77395310674106
// MI455X (gfx1250) — compile-verified
//
#include <hip/hip_runtime.h>
#include <math.h>

// ---------------- types for WMMA ----------------
typedef float v2f __attribute__((ext_vector_type(2)));
typedef float v8f __attribute__((ext_vector_type(8)));

#define EPSF 1e-12f

// =====================================================================
// Batched NT GEMM with optional row scaling on both operands.
// C[z][m][n] = sa[z][m]*sb[z][n] * sum_k A[z][m][k] * B[z][n][k]
// Requires: M,N multiples of 128; K multiple of 16; 16B-aligned rows.
// Block: 256 threads = 8 waves (2 M x 4 N); block tile 128x128;
// wave tile 64x32 = 4x2 WMMA tiles -> 6 LDS frag loads per 8 WMMAs and
// 8 independent accumulator chains (hides V_WMMA_F32_16X16X4_F32 latency).
// =====================================================================
__global__ __launch_bounds__(256) void dha_gemm_nt_wmma(
    const float* __restrict__ A, const float* __restrict__ B,
    float* __restrict__ C, int K, int lda, int ldb, int ldc,
    const float* __restrict__ sa, const float* __restrict__ sb,
    long long strideA, long long strideB, long long strideC, int strideS)
{
  constexpr int KT = 16;
  __shared__ float la[128][KT + 1];
  __shared__ float lb[128][KT + 1];

  const int z = blockIdx.z;
  A += (size_t)z * strideA;
  B += (size_t)z * strideB;
  C += (size_t)z * strideC;
  if (sa) sa += (size_t)z * strideS;
  if (sb) sb += (size_t)z * strideS;

  const int tid  = threadIdx.x;
  const int lane = tid & 31;
  const int wave = tid >> 5;       // 0..7
  const int wm   = wave >> 2;      // 0..1 : M offset wm*64
  const int wn   = wave & 3;       // 0..3 : N offset wn*32
  const int g    = lane >> 4;      // 0..1
  const int l15  = lane & 15;
  const int m0   = blockIdx.y * 128;
  const int n0   = blockIdx.x * 128;

  // cooperative loads: thread covers rows (ldr, ldr+64), 4 floats each
  const int ldr = tid >> 2;          // 0..63
  const int ldk = (tid & 3) * 4;     // 0,4,8,12

  v8f acc[4][2];
  #pragma unroll
  for (int t = 0; t < 4; ++t)
    #pragma unroll
    for (int u = 0; u < 2; ++u) acc[t][u] = (v8f){};

  const float sA0 = sa ? sa[m0 + ldr]      : 1.0f;
  const float sA1 = sa ? sa[m0 + ldr + 64] : 1.0f;
  const float sB0 = sb ? sb[n0 + ldr]      : 1.0f;
  const float sB1 = sb ? sb[n0 + ldr + 64] : 1.0f;

  for (int k0 = 0; k0 < K; k0 += KT) {
    float4 a0 = *(const float4*)(A + (size_t)(m0 + ldr)      * lda + k0 + ldk);
    float4 a1 = *(const float4*)(A + (size_t)(m0 + ldr + 64) * lda + k0 + ldk);
    float4 b0 = *(const float4*)(B + (size_t)(n0 + ldr)      * ldb + k0 + ldk);
    float4 b1 = *(const float4*)(B + (size_t)(n0 + ldr + 64) * ldb + k0 + ldk);
    la[ldr][ldk + 0] = a0.x * sA0; la[ldr][ldk + 1] = a0.y * sA0;
    la[ldr][ldk + 2] = a0.z * sA0; la[ldr][ldk + 3] = a0.w * sA0;
    la[ldr + 64][ldk + 0] = a1.x * sA1; la[ldr + 64][ldk + 1] = a1.y * sA1;
    la[ldr + 64][ldk + 2] = a1.z * sA1; la[ldr + 64][ldk + 3] = a1.w * sA1;
    lb[ldr][ldk + 0] = b0.x * sB0; lb[ldr][ldk + 1] = b0.y * sB0;
    lb[ldr][ldk + 2] = b0.z * sB0; lb[ldr][ldk + 3] = b0.w * sB0;
    lb[ldr + 64][ldk + 0] = b1.x * sB1; lb[ldr + 64][ldk + 1] = b1.y * sB1;
    lb[ldr + 64][ldk + 2] = b1.z * sB1; lb[ldr + 64][ldk + 3] = b1.w * sB1;
    __syncthreads();

    #pragma unroll
    for (int kk = 0; kk < KT; kk += 4) {
      // A frag 16x4: lanes 0-15 hold K={kk,kk+1}, lanes 16-31 K={kk+2,kk+3}
      v2f af[4], bf[2];
      #pragma unroll
      for (int t = 0; t < 4; ++t) {
        const int ar = wm * 64 + t * 16 + l15;
        af[t].x = la[ar][kk + 2 * g];
        af[t].y = la[ar][kk + 2 * g + 1];
      }
      #pragma unroll
      for (int u = 0; u < 2; ++u) {
        const int bc = wn * 32 + u * 16 + l15;
        bf[u].x = lb[bc][kk + 2 * g];
        bf[u].y = lb[bc][kk + 2 * g + 1];
      }
      #pragma unroll
      for (int t = 0; t < 4; ++t)
        #pragma unroll
        for (int u = 0; u < 2; ++u)
          acc[t][u] = __builtin_amdgcn_wmma_f32_16x16x4_f32(
              false, af[t], false, bf[u], (short)0, acc[t][u], false, false);
    }
    __syncthreads();
  }

  // C/D 16x16 layout: lane L -> N = L&15, M = r + 8*(L>>4) for VGPR r
  #pragma unroll
  for (int t = 0; t < 4; ++t)
    #pragma unroll
    for (int u = 0; u < 2; ++u)
      #pragma unroll
      for (int r = 0; r < 8; ++r) {
        const int row = m0 + wm * 64 + t * 16 + r + 8 * g;
        const int col = n0 + wn * 32 + u * 16 + l15;
        C[(size_t)row * ldc + col] = acc[t][u][r];
      }
}

// =====================================================================
// Per-row inverse L2 norm: inv[row] = 1/max(||X[row]||, eps)
// =====================================================================
__global__ __launch_bounds__(256) void dha_rownorm_inv(
    const float* __restrict__ X, float* __restrict__ inv, int dim)
{
  __shared__ float red[256];
  const int row = blockIdx.x;
  const float* p = X + (size_t)row * dim;
  float s = 0.0f;
  for (int j = threadIdx.x; j < dim; j += 256) { float v = p[j]; s += v * v; }
  red[threadIdx.x] = s; __syncthreads();
  for (int o = 128; o > 0; o >>= 1) {
    if (threadIdx.x < o) red[threadIdx.x] += red[threadIdx.x + o];
    __syncthreads();
  }
  if (threadIdx.x == 0) inv[row] = 1.0f / fmaxf(sqrtf(red[0]), EPSF);
}

// =====================================================================
// Region centroids of T (8 regions x 1024 points x 2)
// =====================================================================
__global__ __launch_bounds__(256) void dha_centers(
    const float* __restrict__ T, float* __restrict__ centers)
{
  __shared__ float rx[256], ry[256];
  const int v = blockIdx.x;
  float sx = 0.0f, sy = 0.0f;
  for (int i = threadIdx.x; i < 1024; i += 256) {
    sx += T[(size_t)(v * 1024 + i) * 2 + 0];
    sy += T[(size_t)(v * 1024 + i) * 2 + 1];
  }
  rx[threadIdx.x] = sx; ry[threadIdx.x] = sy; __syncthreads();
  for (int o = 128; o > 0; o >>= 1) {
    if (threadIdx.x < o) { rx[threadIdx.x] += rx[threadIdx.x + o];
                           ry[threadIdx.x] += ry[threadIdx.x + o]; }
    __syncthreads();
  }
  if (threadIdx.x == 0) {
    centers[v * 2 + 0] = rx[0] / 1024.0f;
    centers[v * 2 + 1] = ry[0] / 1024.0f;
  }
}

// st_mask[v][m] = dist(center_v, S_m) < 150 ; cnt[v] = #true
__global__ void dha_mask(const float* __restrict__ centers,
                         const float* __restrict__ S,
                         int* __restrict__ mask, int* __restrict__ cnt)
{
  const int t = blockIdx.x * blockDim.x + threadIdx.x;
  if (t >= 8 * 2048) return;
  const int v = t >> 11, m = t & 2047;
  const float dx = centers[v * 2 + 0] - S[m * 2 + 0];
  const float dy = centers[v * 2 + 1] - S[m * 2 + 1];
  const int in = (sqrtf(dx * dx + dy * dy) < 150.0f) ? 1 : 0;
  mask[t] = in;
  if (in) atomicAdd(&cnt[v], 1);
}

// =====================================================================
// Batched top-5 per row (region = blockIdx.y). Reads the symmetric
// element sim[j*n + e] for coalescing. Ties break toward lower index
// (strict >), matching jax.lax.top_k. mask: per-region row of n ints.
// =====================================================================
__global__ void dha_topk5(const float* __restrict__ sim, int n,
                          long long simStride,
                          const int* __restrict__ mask,
                          int* __restrict__ idx)
{
  const int r = blockIdx.y;
  sim += (size_t)r * simStride;
  const int* mk = mask ? mask + (size_t)r * n : nullptr;
  idx += (size_t)r * n * 5;

  const int e = blockIdx.x * blockDim.x + threadIdx.x;
  if (e >= n) return;
  if (mk && !mk[e]) {
    #pragma unroll
    for (int k = 0; k < 5; ++k) idx[e * 5 + k] = -1;
    return;
  }
  float tv[5]; int ti[5];
  #pragma unroll
  for (int k = 0; k < 5; ++k) { tv[k] = -3.4e38f; ti[k] = -1; }
  for (int j = 0; j < n; ++j) {
    if (mk && !mk[j]) continue;
    const float v = sim[(size_t)j * n + e];   // symmetric element, coalesced
    if (v > tv[4]) {
      int p = 4;
      while (p > 0 && v > tv[p - 1]) { tv[p] = tv[p - 1]; ti[p] = ti[p - 1]; --p; }
      tv[p] = v; ti[p] = j;
    }
  }
  #pragma unroll
  for (int k = 0; k < 5; ++k) idx[e * 5 + k] = ti[k];
}

// node degrees, batched: idx8 is 8 regions x n x 5 (contiguous);
// integer atomics => order-independent, deterministic
__global__ void dha_degree(const int* __restrict__ idx8, int total, int n5,
                           int n, int* __restrict__ dv8)
{
  const int t = blockIdx.x * blockDim.x + threadIdx.x;
  if (t >= total) return;
  const int reg = t / n5;
  const int m = idx8[t];
  if (m >= 0) atomicAdd(&dv8[(size_t)reg * n + m], 1);
}

__global__ void dha_dinvhalf(const int* __restrict__ dv8,
                             float* __restrict__ dih8, int total)
{
  const int i = blockIdx.x * blockDim.x + threadIdx.x;
  if (i < total) dih8[i] = 1.0f / sqrtf(fmaxf((float)dv8[i], EPSF));
}

// per-edge aggregate, batched: G[r][e] = (1/de) sum_{j in top5(r,e)} dih[r][j]*Xw[r][j]
__global__ __launch_bounds__(256) void dha_edge_agg(
    const int* __restrict__ idx8, const float* __restrict__ dih8,
    const float* __restrict__ Xw, long long XwStride,
    float* __restrict__ G8, int n)
{
  const int e = blockIdx.x, r = blockIdx.y, d = threadIdx.x;
  const int*   idx = idx8 + (size_t)r * n * 5;
  const float* dih = dih8 + (size_t)r * n;
  const float* xw  = Xw   + (size_t)r * XwStride;
  float*       G   = G8   + (size_t)r * n * 256;

  int m[5];
  #pragma unroll
  for (int k = 0; k < 5; ++k) m[k] = idx[e * 5 + k];
  float g = 0.0f, de = 0.0f;
  #pragma unroll
  for (int k = 0; k < 5; ++k)
    if (m[k] >= 0) { g += dih[m[k]] * xw[(size_t)m[k] * 256 + d]; de += 1.0f; }
  G[(size_t)e * 256 + d] = (de > 0.0f) ? g / de : 0.0f;
}

// gather per node (fixed e-order => deterministic), then exact gelu:
// Xnew[r][i] = gelu( dih[r][i] * sum_{e : i in top5(r,e)} G[r][e] )
__global__ __launch_bounds__(256) void dha_gather_gelu(
    const int* __restrict__ idx8, const float* __restrict__ dih8,
    const float* __restrict__ G8, float* __restrict__ Xn8, int n)
{
  const int i = blockIdx.x, r = blockIdx.y, d = threadIdx.x;
  const int*   idx = idx8 + (size_t)r * n * 5;
  const float* dih = dih8 + (size_t)r * n;
  const float* G   = G8   + (size_t)r * n * 256;
  float*       Xn  = Xn8  + (size_t)r * n * 256;

  float y = 0.0f;
  for (int e = 0; e < n; ++e) {
    const int* me = idx + (size_t)e * 5;
    const bool mem = (me[0] == i) | (me[1] == i) | (me[2] == i) |
                     (me[3] == i) | (me[4] == i);
    if (mem) y += G[(size_t)e * 256 + d];
  }
  const float x = dih[i] * y;
  Xn[(size_t)i * 256 + d] = 0.5f * x * (1.0f + erff(x * 0.70710678118654752f));
}

// hierarchical mean-pool stage 1 (batched): partial sums over row blocks
__global__ __launch_bounds__(256) void dha_pool_partial(
    const float* __restrict__ Xn8, float* __restrict__ Pp8, int n)
{
  const int b = blockIdx.x, r = blockIdx.y, d = threadIdx.x;
  const float* Xn = Xn8 + (size_t)r * n * 256;
  const int rows = n >> 4;           // 16 blocks per region
  float s = 0.0f;
  const int i0 = b * rows;
  for (int q = 0; q < rows; ++q) s += Xn[(size_t)(i0 + q) * 256 + d];
  Pp8[((size_t)r * 16 + b) * 256 + d] = s;
}

// stage 2 (batched): z[r][d] = (sum of partials) / count
__global__ __launch_bounds__(256) void dha_pool_final(
    const float* __restrict__ Pp8, float* __restrict__ z,
    const int* __restrict__ cnt_i, float cnt_f)
{
  const int r = blockIdx.x, d = threadIdx.x;
  float s = 0.0f;
  for (int b = 0; b < 16; ++b) s += Pp8[((size_t)r * 16 + b) * 256 + d];
  const float c = cnt_i ? (float)cnt_i[r] : cnt_f;
  z[(size_t)r * 256 + d] = s / c;
}

// =====================================================================
// Final symmetric InfoNCE loss over 8x8 logits (single block).
// =====================================================================
__global__ __launch_bounds__(256) void dha_loss(
    const float* __restrict__ zhe, const float* __restrict__ zst,
    float* __restrict__ out)
{
  __shared__ float nh[8], ns[8];
  __shared__ float logits[8][8];
  __shared__ float red[256];
  const int t = threadIdx.x;
  for (int v = 0; v < 8; ++v) {
    float x = zhe[v * 256 + t]; red[t] = x * x; __syncthreads();
    for (int o = 128; o > 0; o >>= 1) { if (t < o) red[t] += red[t + o]; __syncthreads(); }
    if (t == 0) nh[v] = fmaxf(sqrtf(red[0]), EPSF);
    __syncthreads();
    x = zst[v * 256 + t]; red[t] = x * x; __syncthreads();
    for (int o = 128; o > 0; o >>= 1) { if (t < o) red[t] += red[t + o]; __syncthreads(); }
    if (t == 0) ns[v] = fmaxf(sqrtf(red[0]), EPSF);
    __syncthreads();
  }
  if (t < 64) {
    const int v = t >> 3, u = t & 7;
    float dot = 0.0f;
    for (int d = 0; d < 256; ++d) dot += zhe[v * 256 + d] * zst[u * 256 + d];
    logits[v][u] = dot / (nh[v] * ns[u]) * (1.0f / 0.07f);
  }
  __syncthreads();
  if (t == 0) {
    float l1 = 0.0f, l2 = 0.0f;
    for (int v = 0; v < 8; ++v) {
      float mx = -3.4e38f;
      for (int u = 0; u < 8; ++u) mx = fmaxf(mx, logits[v][u]);
      float se = 0.0f;
      for (int u = 0; u < 8; ++u) se += expf(logits[v][u] - mx);
      l1 += (logf(se) + mx) - logits[v][v];
      mx = -3.4e38f;
      for (int u = 0; u < 8; ++u) mx = fmaxf(mx, logits[u][v]);
      se = 0.0f;
      for (int u = 0; u < 8; ++u) se += expf(logits[u][v] - mx);
      l2 += (logf(se) + mx) - logits[v][v];
    }
    out[0] = 0.5f * (l1 / 8.0f + l2 / 8.0f);
  }
}

// =====================================================================
// Host orchestration
// =====================================================================
extern "C" void kernel_launch(void* const* d_in, const int* in_sizes, int n_in,
                              void* d_out, int out_size, void* d_ws, size_t ws_size,
                              hipStream_t stream)
{
  (void)in_sizes; (void)n_in; (void)out_size; (void)ws_size;
  const float* F    = (const float*)d_in[0];   // 8192 x 1024
  const float* T    = (const float*)d_in[1];   // 8192 x 2
  /* d_in[2] region_indices: implicit (contiguous equal blocks) */
  const float* E    = (const float*)d_in[3];   // 2048 x 512
  const float* S    = (const float*)d_in[4];   // 2048 x 2
  const float* W_he = (const float*)d_in[5];   // 256 x 1024
  const float* W_st = (const float*)d_in[6];   // 256 x 512
  float* out = (float*)d_out;

  char* p = (char*)d_ws;
  auto alloc = [&](size_t bytes) -> void* {
    void* q = (void*)p; p += (bytes + 255) & ~(size_t)255; return q;
  };
  float* invF    = (float*)alloc(8192 * 4);
  float* invE    = (float*)alloc(2048 * 4);
  float* centers = (float*)alloc(8 * 2 * 4);
  int*   cnt     = (int*)alloc(8 * 4);
  int*   mask    = (int*)alloc(8 * 2048 * 4);
  float* XwHe    = (float*)alloc((size_t)8192 * 256 * 4);
  float* XwSt    = (float*)alloc((size_t)2048 * 256 * 4);
  float* simSt   = (float*)alloc((size_t)2048 * 2048 * 4);
  float* simHe8  = (float*)alloc((size_t)8 * 1024 * 1024 * 4);  // all 8 regions
  int*   idxHe8  = (int*)alloc((size_t)8 * 1024 * 5 * 4);
  int*   idxSt8  = (int*)alloc((size_t)8 * 2048 * 5 * 4);
  int*   dv8     = (int*)alloc((size_t)8 * 2048 * 4);           // reused HE/ST
  float* dih8    = (float*)alloc((size_t)8 * 2048 * 4);         // reused HE/ST
  float* G8      = (float*)alloc((size_t)8 * 2048 * 256 * 4);   // reused HE/ST
  float* Xn8     = (float*)alloc((size_t)8 * 2048 * 256 * 4);   // reused HE/ST
  float* Pp8     = (float*)alloc((size_t)8 * 16 * 256 * 4);
  float* zhe     = (float*)alloc(8 * 256 * 4);
  float* zst     = (float*)alloc(8 * 256 * 4);

  const dim3 blk(256);

  // normalization scales + geometry
  dha_rownorm_inv<<<8192, blk, 0, stream>>>(F, invF, 1024);
  dha_rownorm_inv<<<2048, blk, 0, stream>>>(E, invE, 512);
  dha_centers<<<8, blk, 0, stream>>>(T, centers);
  hipMemsetAsync(cnt, 0, 8 * 4, stream);
  dha_mask<<<(8 * 2048) / 256, blk, 0, stream>>>(centers, S, mask, cnt);

  // projections (WMMA): Xw_he = F @ W_he^T, Xw_st = E @ W_st^T
  dha_gemm_nt_wmma<<<dim3(256 / 128, 8192 / 128, 1), blk, 0, stream>>>(
      F, W_he, XwHe, 1024, 1024, 1024, 256, nullptr, nullptr, 0, 0, 0, 0);
  dha_gemm_nt_wmma<<<dim3(256 / 128, 2048 / 128, 1), blk, 0, stream>>>(
      E, W_st, XwSt, 512, 512, 512, 256, nullptr, nullptr, 0, 0, 0, 0);
  // ST similarity: computed ONCE (mask varies per region, sim does not)
  dha_gemm_nt_wmma<<<dim3(2048 / 128, 2048 / 128, 1), blk, 0, stream>>>(
      E, E, simSt, 512, 512, 512, 2048, invE, invE, 0, 0, 0, 0);
  // HE similarities: all 8 regions in one batched launch (blockIdx.z)
  dha_gemm_nt_wmma<<<dim3(1024 / 128, 1024 / 128, 8), blk, 0, stream>>>(
      F, F, simHe8, 1024, 1024, 1024, 1024, invF, invF,
      (long long)1024 * 1024, (long long)1024 * 1024,
      (long long)1024 * 1024, 1024);

  // ---- HE side: batched over 8 regions ----
  dha_topk5<<<dim3(4, 8), blk, 0, stream>>>(
      simHe8, 1024, (long long)1024 * 1024, nullptr, idxHe8);
  hipMemsetAsync(dv8, 0, (size_t)8 * 1024 * 4, stream);
  dha_degree<<<(8 * 1024 * 5 + 255) / 256, blk, 0, stream>>>(
      idxHe8, 8 * 1024 * 5, 1024 * 5, 1024, dv8);
  dha_dinvhalf<<<(8 * 1024 + 255) / 256, blk, 0, stream>>>(dv8, dih8, 8 * 1024);
  dha_edge_agg<<<dim3(1024, 8), blk, 0, stream>>>(
      idxHe8, dih8, XwHe, (long long)1024 * 256, G8, 1024);
  dha_gather_gelu<<<dim3(1024, 8), blk, 0, stream>>>(idxHe8, dih8, G8, Xn8, 1024);
  dha_pool_partial<<<dim3(16, 8), blk, 0, stream>>>(Xn8, Pp8, 1024);
  dha_pool_final<<<8, blk, 0, stream>>>(Pp8, zhe, nullptr, 1024.0f);

  // ---- ST side: shared spots, per-region radius mask, batched ----
  dha_topk5<<<dim3(8, 8), blk, 0, stream>>>(simSt, 2048, 0, mask, idxSt8);
  hipMemsetAsync(dv8, 0, (size_t)8 * 2048 * 4, stream);
  dha_degree<<<(8 * 2048 * 5 + 255) / 256, blk, 0, stream>>>(
      idxSt8, 8 * 2048 * 5, 2048 * 5, 2048, dv8);
  dha_dinvhalf<<<(8 * 2048 + 255) / 256, blk, 0, stream>>>(dv8, dih8, 8 * 2048);
  dha_edge_agg<<<dim3(2048, 8), blk, 0, stream>>>(
      idxSt8, dih8, XwSt, 0, G8, 2048);
  dha_gather_gelu<<<dim3(2048, 8), blk, 0, stream>>>(idxSt8, dih8, G8, Xn8, 2048);
  dha_pool_partial<<<dim3(16, 8), blk, 0, stream>>>(Xn8, Pp8, 2048);
  dha_pool_final<<<8, blk, 0, stream>>>(Pp8, zst, cnt, 0.0f);

  dha_loss<<<1, blk, 0, stream>>>(zhe, zst, out);
}